// myGAT_40570261078602
// MI455X (gfx1250) — compile-verified
//
#include <hip/hip_runtime.h>

typedef float v2f __attribute__((ext_vector_type(2)));
typedef float v8f __attribute__((ext_vector_type(8)));

#define GN 100000
#define GE 1600000
#define GD 128

// ---------------------------------------------------------------------------
// Degree kernels
// ---------------------------------------------------------------------------
__global__ void deg_count_kernel(const int* __restrict__ dst, float* __restrict__ deg, int E) {
    int e = blockIdx.x * blockDim.x + threadIdx.x;
    if (e < E) atomicAdd(&deg[dst[e]], 1.0f);
}

__global__ void invdeg_kernel(float* __restrict__ deg, int n) {
    int i = blockIdx.x * blockDim.x + threadIdx.x;
    if (i < n) {
        float d = deg[i];
        deg[i] = 1.0f / fmaxf(d, 1.0f);
    }
}

// ---------------------------------------------------------------------------
// Edge scatter: neigh[dst] += h[src]   (one wave32 per edge, float4 per lane,
// 4 x global_atomic_add_f32 per lane; h (51 MB) is L2-resident on MI455X)
// ---------------------------------------------------------------------------
__global__ void scatter_kernel(const float* __restrict__ h,
                               const int* __restrict__ src,
                               const int* __restrict__ dst,
                               float* __restrict__ neigh, int E) {
    int lane = threadIdx.x & 31;
    int wave = threadIdx.x >> 5;
    int e = blockIdx.x * (blockDim.x >> 5) + wave;
    if (e >= E) return;
    int s = src[e];
    int d = dst[e];
    const float4* hp = (const float4*)(h + (size_t)s * GD);
    float4 v = hp[lane];
    float* np = neigh + (size_t)d * GD + lane * 4;
    atomicAdd(np + 0, v.x);
    atomicAdd(np + 1, v.y);
    atomicAdd(np + 2, v.z);
    atomicAdd(np + 3, v.w);
}

// ---------------------------------------------------------------------------
// LDS-staged WMMA fp32 GEMM for Dout == 128.
// One workgroup (256 threads = 8 wave32) per 16-row m-tile; wave w computes
// n-tile w. The 16xK A-tile (contiguous rows in memory) is staged into LDS:
//   FUSE=false: pure copy -> GLOBAL_LOAD_ASYNC_TO_LDS_B128 (ASYNCcnt DMA)
//   FUSE=true : Aeff = X + neigh*inv_deg computed in VALU, ds_store_b128
// Row pad of +4 floats (stride 132 dwords) avoids LDS bank conflicts on the
// per-lane ds_load_b64 A-fragment reads.
// ---------------------------------------------------------------------------
template <int K, bool FUSE, bool SIG>
__global__ __launch_bounds__(256) void gemm_lds_wmma_kernel(
        const float* __restrict__ X,
        const float* __restrict__ neigh,
        const float* __restrict__ invdeg,
        const float* __restrict__ W,
        const float* __restrict__ bias,
        float* __restrict__ Y) {
    constexpr int KP = K + 4;                  // padded row stride (floats)
    __shared__ float sA[16 * KP];              // single LDS object -> frame offset 0

    const int tid = threadIdx.x;
    const int mt = blockIdx.x;
    const size_t base = (size_t)mt * 16 * K;   // 16 consecutive rows, contiguous
    constexpr int NVEC = 16 * K / 4;           // float4 elements (multiple of 256)

    if constexpr (FUSE) {
        for (int v = tid; v < NVEC; v += 256) {
            const int linear = v * 4;
            const int row = linear / K;
            const int colo = linear - row * K;
            float4 x = *(const float4*)(X + base + linear);
            float4 nn = *(const float4*)(neigh + base + linear);
            const float idg = invdeg[mt * 16 + row];
            float4 r;
            r.x = x.x + nn.x * idg;
            r.y = x.y + nn.y * idg;
            r.z = x.z + nn.z * idg;
            r.w = x.w + nn.w * idg;
            *(float4*)(&sA[row * KP + colo]) = r;    // ds_store_b128
        }
        __syncthreads();
    } else {
        for (int v = tid; v < NVEC; v += 256) {
            const int linear = v * 4;
            const int row = linear / K;
            const int colo = linear - row * K;
            unsigned ldsoff = (unsigned)((row * KP + colo) * 4);
            unsigned long long gaddr = (unsigned long long)(X + base + linear);
            // per-lane 16B global -> LDS DMA, tracked by ASYNCcnt
            asm volatile("global_load_async_to_lds_b128 %0, %1, off"
                         :: "v"(ldsoff), "v"(gaddr) : "memory");
        }
        asm volatile("s_wait_asynccnt 0" ::: "memory");
        __syncthreads();
    }

    // ---- compute: wave w -> columns [w*16, w*16+16) ----
    const int lane = tid & 31;
    const int wave = tid >> 5;
    const int half = lane >> 4;                // K sub-pair selector (A/B frags)
    const int l15 = lane & 15;
    const int col = wave * 16 + l15;
    const float* arow = &sA[l15 * KP];

    v8f acc = {0.f, 0.f, 0.f, 0.f, 0.f, 0.f, 0.f, 0.f};

#pragma unroll 4
    for (int k = 0; k < K; k += 4) {
        const int ka = k + half * 2;
        v2f a = *(const v2f*)(arow + ka);      // ds_load_b64, conflict-free
        v2f b;
        b.x = W[(ka + 0) * GD + col];
        b.y = W[(ka + 1) * GD + col];
        acc = __builtin_amdgcn_wmma_f32_16x16x4_f32(
            false, a, false, b, (short)0, acc, false, false);
    }

    const float bb = bias[col];
    const int rbase = mt * 16 + half * 8;      // C: VGPR r = rows r / r+8
#pragma unroll
    for (int r = 0; r < 8; ++r) {
        float v = acc[r] + bb;
        if (SIG) v = 1.0f / (1.0f + __expf(-v));
        Y[(size_t)(rbase + r) * GD + col] = v;
    }
}

// ---------------------------------------------------------------------------
// Per-wave-tile WMMA GEMM (used for the final layer: Dout=16 -> NT=1, so
// there is no cross-wave A reuse to exploit with LDS staging).
// ---------------------------------------------------------------------------
template <bool FUSE, bool SIG>
__global__ void gemm_wmma_f32_kernel(const float* __restrict__ X,
                                     const float* __restrict__ neigh,
                                     const float* __restrict__ invdeg,
                                     const float* __restrict__ W,
                                     const float* __restrict__ bias,
                                     float* __restrict__ Y,
                                     int M, int K, int Dout) {
    const int lane = threadIdx.x & 31;
    const int wave = threadIdx.x >> 5;
    const int tile = blockIdx.x * (blockDim.x >> 5) + wave;
    const int ntiles = Dout >> 4;
    const int totalTiles = (M >> 4) * ntiles;
    if (tile >= totalTiles) return;            // wave-uniform: EXEC all-1s below

    const int mt = tile / ntiles;
    const int nt = tile - mt * ntiles;
    const int half = lane >> 4;
    const int l15 = lane & 15;
    const int arow = mt * 16 + l15;
    const int col = nt * 16 + l15;

    const float* xrow = X + (size_t)arow * K;
    const float* nrow = FUSE ? (neigh + (size_t)arow * K) : nullptr;
    const float idg = FUSE ? invdeg[arow] : 0.0f;

    v8f acc = {0.f, 0.f, 0.f, 0.f, 0.f, 0.f, 0.f, 0.f};

    for (int k = 0; k < K; k += 4) {
        const int ka = k + half * 2;
        v2f a, b;
        a.x = xrow[ka + 0];
        a.y = xrow[ka + 1];
        if (FUSE) {
            a.x += nrow[ka + 0] * idg;
            a.y += nrow[ka + 1] * idg;
        }
        b.x = W[(size_t)(ka + 0) * Dout + col];
        b.y = W[(size_t)(ka + 1) * Dout + col];
        acc = __builtin_amdgcn_wmma_f32_16x16x4_f32(
            false, a, false, b, (short)0, acc, false, false);
    }

    const float bb = bias[col];
    const int rbase = mt * 16 + half * 8;
#pragma unroll
    for (int r = 0; r < 8; ++r) {
        float v = acc[r] + bb;
        if (SIG) v = 1.0f / (1.0f + __expf(-v));
        Y[(size_t)(rbase + r) * Dout + col] = v;
    }
}

// ---------------------------------------------------------------------------
// Launch
// ---------------------------------------------------------------------------
extern "C" void kernel_launch(void* const* d_in, const int* in_sizes, int n_in,
                              void* d_out, int out_size, void* d_ws, size_t ws_size,
                              hipStream_t stream) {
    const float* feat0 = (const float*)d_in[0];   // 60000 x 512
    const float* feat1 = (const float*)d_in[1];   // 40000 x 256
    const int*   src   = (const int*)d_in[2];     // E
    const int*   dst   = (const int*)d_in[3];     // E
    const float* W_fc0 = (const float*)d_in[4];   // 512 x 128
    const float* b_fc0 = (const float*)d_in[5];
    const float* W_fc1 = (const float*)d_in[6];   // 256 x 128
    const float* b_fc1 = (const float*)d_in[7];
    const float* W_g0  = (const float*)d_in[8];   // 128 x 128
    const float* b_g0  = (const float*)d_in[9];
    const float* W_g1  = (const float*)d_in[10];  // 128 x 128
    const float* b_g1  = (const float*)d_in[11];
    const float* W_out = (const float*)d_in[12];  // 128 x 16
    const float* b_out = (const float*)d_in[13];
    float* out = (float*)d_out;                   // 100000 x 16

    float* ws = (float*)d_ws;
    float* invdeg = ws;                           // N floats
    float* h      = ws + GN;                      // N*D
    float* nb     = h + (size_t)GN * GD;          // N*D (scatter accumulator)
    float* h2     = nb + (size_t)GN * GD;         // N*D (ping-pong)

    // ---- degree -> inv_deg ----
    hipMemsetAsync(invdeg, 0, (size_t)GN * sizeof(float), stream);
    deg_count_kernel<<<(GE + 255) / 256, 256, 0, stream>>>(dst, invdeg, GE);
    invdeg_kernel<<<(GN + 255) / 256, 256, 0, stream>>>(invdeg, GN);

    // ---- per-type input projections -> h (no activation, async LDS staging) ----
    gemm_lds_wmma_kernel<512, false, false><<<60000 / 16, 256, 0, stream>>>(
        feat0, nullptr, nullptr, W_fc0, b_fc0, h);
    gemm_lds_wmma_kernel<256, false, false><<<40000 / 16, 256, 0, stream>>>(
        feat1, nullptr, nullptr, W_fc1, b_fc1, h + (size_t)60000 * GD);

    const int scatterWaves = 8;                   // 256 threads = 8 edges/block
    const int scatterGrid = (GE + scatterWaves - 1) / scatterWaves;

    // ---- GIN layer 0: h -> h2 ----
    hipMemsetAsync(nb, 0, (size_t)GN * GD * sizeof(float), stream);
    scatter_kernel<<<scatterGrid, scatterWaves * 32, 0, stream>>>(h, src, dst, nb, GE);
    gemm_lds_wmma_kernel<128, true, true><<<GN / 16, 256, 0, stream>>>(
        h, nb, invdeg, W_g0, b_g0, h2);

    // ---- GIN layer 1: h2 -> h ----
    hipMemsetAsync(nb, 0, (size_t)GN * GD * sizeof(float), stream);
    scatter_kernel<<<scatterGrid, scatterWaves * 32, 0, stream>>>(h2, src, dst, nb, GE);
    gemm_lds_wmma_kernel<128, true, true><<<GN / 16, 256, 0, stream>>>(
        h2, nb, invdeg, W_g1, b_g1, h);

    // ---- output GIN layer: h -> out (100000 x 16) ----
    hipMemsetAsync(nb, 0, (size_t)GN * GD * sizeof(float), stream);
    scatter_kernel<<<scatterGrid, scatterWaves * 32, 0, stream>>>(h, src, dst, nb, GE);
    {
        const int tiles = (GN / 16) * (16 / 16);
        const int grid = (tiles + 3) / 4;
        gemm_wmma_f32_kernel<true, true><<<grid, 128, 0, stream>>>(
            h, nb, invdeg, W_out, b_out, out, GN, GD, 16);
    }
}